// CAICalculator_12206297055790
// MI455X (gfx1250) — compile-verified
//
#include <hip/hip_runtime.h>
#include <math.h>

typedef float v2f __attribute__((ext_vector_type(2)));
typedef float v8f __attribute__((ext_vector_type(8)));
typedef int   v4i __attribute__((ext_vector_type(4)));

#if defined(__has_builtin)
#if __has_builtin(__builtin_amdgcn_global_load_async_to_lds_b32)
#define HAVE_ASYNC_LDS 1
#endif
#if __has_builtin(__builtin_amdgcn_s_wait_asynccnt)
#define HAVE_WAIT_ASYNC 1
#endif
#endif
#ifndef HAVE_ASYNC_LDS
#define HAVE_ASYNC_LDS 0
#endif
#ifndef HAVE_WAIT_ASYNC
#define HAVE_WAIT_ASYNC 0
#endif

// One block (8 wave32s) per row. Streams L codons (int32) + L mask bytes,
// looks up log-weights in a 64-entry LDS table (1 entry per bank -> conflict
// free), reduces with shfl_xor + a single v_wmma_f32_16x16x4_f32 (B = ones).
__global__ __launch_bounds__(256)
void cai_row_kernel(const float* __restrict__ weight,       // [S,64]
                    const int* __restrict__ codon,          // [B,L]
                    const int* __restrict__ species,        // [B]
                    const unsigned char* __restrict__ mask, // [B,L]
                    float* __restrict__ out,                // [B]
                    int L) {
    __shared__ float wtab[64];   // log-weight table for this row's species
    __shared__ float sred[16];   // [0..7]=acc partials, [8..15]=cnt partials

    const int tid  = threadIdx.x;
    const int row  = blockIdx.x;
    const int lane = tid & 31;
    const int wave = tid >> 5;
    const int s    = species[row];

    // ---- Stage the 64-float weight row into LDS (async-to-LDS if available),
    //      then convert in place to log(clip(w, 1e-8)).
#if HAVE_ASYNC_LDS
    if (tid < 64) {
        __attribute__((address_space(1))) int* g =
            (__attribute__((address_space(1))) int*)(weight + s * 64 + tid);
        __attribute__((address_space(3))) int* l =
            (__attribute__((address_space(3))) int*)&wtab[tid];
        __builtin_amdgcn_global_load_async_to_lds_b32(g, l, 0, 0);
    }
#if HAVE_WAIT_ASYNC
    __builtin_amdgcn_s_wait_asynccnt(0);
#else
    asm volatile("s_wait_asynccnt 0" ::: "memory");
#endif
    __syncthreads();
    if (tid < 64) wtab[tid] = logf(fmaxf(wtab[tid], 1e-8f));
#else
    if (tid < 64) wtab[tid] = logf(fmaxf(weight[s * 64 + tid], 1e-8f));
#endif
    __syncthreads();

    // ---- Streaming phase: 4-codon chunks (b128) + uint32 (4 bytes) of mask.
    const size_t base = (size_t)row * (size_t)L;
    const v4i* __restrict__ crow = (const v4i*)(codon + base);
    const unsigned int* __restrict__ mrow = (const unsigned int*)(mask + base);
    const int nvec = L >> 2;

    float acc = 0.0f, cnt = 0.0f;
    for (int v = tid; v < nvec; v += 256) {
        v4i c          = __builtin_nontemporal_load(crow + v);
        unsigned int m = __builtin_nontemporal_load(mrow + v);
        float m0 = (m & 0xffu)         ? 1.0f : 0.0f;
        float m1 = ((m >> 8) & 0xffu)  ? 1.0f : 0.0f;
        float m2 = ((m >> 16) & 0xffu) ? 1.0f : 0.0f;
        float m3 = (m >> 24)           ? 1.0f : 0.0f;
        acc += wtab[c.x & 63] * m0;  cnt += m0;
        acc += wtab[c.y & 63] * m1;  cnt += m1;
        acc += wtab[c.z & 63] * m2;  cnt += m2;
        acc += wtab[c.w & 63] * m3;  cnt += m3;
    }

    // ---- Intra-wave reduction (wave32).
    for (int off = 16; off > 0; off >>= 1) {
        acc += __shfl_xor(acc, off, 32);
        cnt += __shfl_xor(cnt, off, 32);
    }
    if (lane == 0) { sred[wave] = acc; sred[8 + wave] = cnt; }
    __syncthreads();

    // ---- Cross-wave reduction via one WMMA (wave 0 only, EXEC all ones).
    // A(16x4): row0 = acc[0..3], row1 = acc[4..7], row2 = cnt[0..3],
    // row3 = cnt[4..7], rows 4..15 zero.  B = ones  =>  D[m][0] = row-sum(A,m).
    // A layout: lane m (m<16): {A[m][0], A[m][1]}; lane 16+m: {A[m][2], A[m][3]}.
    if (wave == 0) {
        const int r   = lane & 15;
        const int kh  = lane >> 4;              // 0 -> k={0,1}, 1 -> k={2,3}
        const int idx = (r & 3) * 4 + kh * 2;   // sred index of A[r][k]
        const float valid = (r < 4) ? 1.0f : 0.0f;
        v2f a; a[0] = sred[idx] * valid; a[1] = sred[idx + 1] * valid;
        v2f b; b[0] = 1.0f; b[1] = 1.0f;
        v8f cm = {};
        v8f d = __builtin_amdgcn_wmma_f32_16x16x4_f32(
            /*neg_a=*/false, a, /*neg_b=*/false, b,
            /*c_mod=*/(short)0, cm, /*reuse_a=*/false, /*reuse_b=*/false);
        if (lane == 0) {
            // lane0: d[0]=D[0][0]=sum acc[0..3], d[1]=D[1][0]=sum acc[4..7],
            //        d[2]=D[2][0]=sum cnt[0..3], d[3]=D[3][0]=sum cnt[4..7]
            float sum_log = d[0] + d[1];
            float nvalid  = fmaxf(d[2] + d[3], 1.0f);
            out[row] = expf(sum_log / nvalid);
        }
    }
}

extern "C" void kernel_launch(void* const* d_in, const int* in_sizes, int n_in,
                              void* d_out, int out_size, void* d_ws, size_t ws_size,
                              hipStream_t stream) {
    const float* weight        = (const float*)d_in[0];        // (S, 64) f32
    const int* codon           = (const int*)d_in[1];          // (B, L) i32
    const int* species         = (const int*)d_in[2];          // (B,)   i32
    const unsigned char* mask  = (const unsigned char*)d_in[3];// (B, L) bool
    float* out                 = (float*)d_out;                // (B,)   f32

    const int B = in_sizes[2];
    const int L = in_sizes[1] / B;

    cai_row_kernel<<<B, 256, 0, stream>>>(weight, codon, species, mask, out, L);
}